// BentPrototypeQuantizer_34359739040
// MI455X (gfx1250) — compile-verified
//
#include <hip/hip_runtime.h>

// CDNA5 / gfx1250, wave32 only.
typedef __attribute__((ext_vector_type(2))) float v2f;
typedef __attribute__((ext_vector_type(8))) float v8f;

#define BLOCK_THREADS   256   // 8 waves
#define WAVES_PER_BLOCK 8
#define TILE_P          16    // points per wave-iteration (one WMMA N-tile)
#define MAX_BLOCKS      1024  // grid-stride: ~8 tiles per wave at P = 1M

// out[p] = codebook[ argmin_k ||x[p]-c_k||^2 ] == codebook[ argmax_k (x.c_k - 0.5||c_k||^2) ]
//
// Transposed mapping: score tile = C(16 codewords x 6) * X^T(6 x 16 points) via
// V_WMMA_F32_16X16X4_F32 (K = 4 + 2 zero-padded), bias -0.5||c||^2 folded into the
// accumulator init. In the D layout a point is a COLUMN: its 64 scores (4 M-tiles
// x 8 row-VGPRs) live in lanes g and g+16 only. Argmax runs as 4 independent
// per-tile compare chains (ILP alongside the WMMAs), a 3-compare combine, and a
// single shfl_xor(16) cross-half exchange.
__global__ __launch_bounds__(BLOCK_THREADS) void bent_quant_wmma_kernel(
    const float* __restrict__ x,    // (P, 6)
    const float* __restrict__ cb,   // (K, 6), K <= 64
    float* __restrict__ out,        // (P, 6)
    int P, int K)
{
    __shared__ float s_cb[64 * 6];
    __shared__ float s_cinit[64];   // -0.5*||c_n||^2 (padding: -1e30, never wins)

    const int tid  = threadIdx.x;
    const int lane = tid & 31;
    const int g    = lane & 15;     // point/column slot within tile
    const int half = lane >> 4;     // 0: rows 0-7 & K-dims 0,1 | 1: rows 8-15 & K-dims 2,3

    if (tid < 64) {
        if (tid < K) {
            const float* c = cb + (size_t)tid * 6;
            float c0=c[0], c1=c[1], c2=c[2], c3=c[3], c4=c[4], c5=c[5];
            s_cb[tid*6+0]=c0; s_cb[tid*6+1]=c1; s_cb[tid*6+2]=c2;
            s_cb[tid*6+3]=c3; s_cb[tid*6+4]=c4; s_cb[tid*6+5]=c5;
            s_cinit[tid] = -0.5f * (c0*c0 + c1*c1 + c2*c2 + c3*c3 + c4*c4 + c5*c5);
        } else {
            s_cb[tid*6+0]=0.f; s_cb[tid*6+1]=0.f; s_cb[tid*6+2]=0.f;
            s_cb[tid*6+3]=0.f; s_cb[tid*6+4]=0.f; s_cb[tid*6+5]=0.f;
            s_cinit[tid] = -1.0e30f;
        }
    }
    __syncthreads();

    // ---- loop-invariant codebook A-fragments + accumulator-init vectors ----
    // A 16x4 f32 layout: lanes 0-15 hold K=0,1 ; lanes 16-31 hold K=2,3 (row M = g).
    v2f a1[4], a2[4];
    v8f cinitv[4];
#pragma unroll
    for (int t = 0; t < 4; ++t) {
        const int n = t * 16 + g;                 // codeword row M for this lane
        const float* cn = s_cb + n * 6;
        a1[t].x = cn[half*2 + 0];                 // half0: d0,d1 | half1: d2,d3
        a1[t].y = cn[half*2 + 1];
        float e4 = cn[4], e5 = cn[5];             // K-dims 4,5 live in lanes 0-15 only
        a2[t].x = half ? 0.f : e4;
        a2[t].y = half ? 0.f : e5;
        const float* ci = s_cinit + t*16 + half*8;  // row bias: M = t*16 + half*8 + r
#pragma unroll
        for (int r = 0; r < 8; ++r) cinitv[t][r] = ci[r];
    }

    const int numTiles   = (P + TILE_P - 1) / TILE_P;
    const int waveGlobal = blockIdx.x * WAVES_PER_BLOCK + (tid >> 5);
    const int waveStride = gridDim.x * WAVES_PER_BLOCK;

    for (int tile = waveGlobal; tile < numTiles; tile += waveStride) {
        const int base = tile * TILE_P;
        int p = base + g;
        if (p >= P) p = P - 1;                    // clamp tail loads; stores guarded
        const float* xp = x + (size_t)p * 6;

        // B 4x16 f32: VGPR0 = rowK0|rowK2, VGPR1 = rowK1|rowK3 across lane halves.
        v2f b1, b2;
        b1.x = xp[half*2 + 0];                    // half0: x0,x1 | half1: x2,x3
        b1.y = xp[half*2 + 1];
        const float* xq = xp + (half ? 0 : 4);    // OOB-safe address select
        float q0 = xq[0], q1 = xq[1];
        b2.x = half ? 0.f : q0;                   // half0: x4,x5 | half1: zero pad
        b2.y = half ? 0.f : q1;

        // 4 independent WMMA chains and 4 independent argmax chains (ILP).
        float tv[4];
        int   ti[4];
#pragma unroll
        for (int t = 0; t < 4; ++t) {
            v8f acc = __builtin_amdgcn_wmma_f32_16x16x4_f32(
                false, a1[t], false, b1, (short)0, cinitv[t], false, false);
            acc = __builtin_amdgcn_wmma_f32_16x16x4_f32(
                false, a2[t], false, b2, (short)0, acc, false, false);
            const int mb = t * 16 + half * 8;     // row index base for this lane
            float v = acc[0];
            int   i = mb;
#pragma unroll
            for (int r = 1; r < 8; ++r) {         // ascending index + strict '>'
                if (acc[r] > v) { v = acc[r]; i = mb + r; }
            }
            tv[t] = v; ti[t] = i;
        }
        // combine the 4 tile results (ascending t keeps smallest index on ties)
        float bv = tv[0];
        int   bi = ti[0];
#pragma unroll
        for (int t = 1; t < 4; ++t) {
            if (tv[t] > bv) { bv = tv[t]; bi = ti[t]; }
        }

        // combine the two lane-halves holding this point's scores
        float ov = __shfl_xor(bv, 16, 32);
        int   oi = __shfl_xor(bi, 16, 32);
        if (ov > bv || (ov == bv && oi < bi)) { bv = ov; bi = oi; }

        // lane g writes dims 0-2, lane g+16 writes dims 3-5 of the winning codeword
        if (base + g < P) {
            const float* c = s_cb + bi * 6 + half * 3;
            float* o = out + (size_t)(base + g) * 6 + half * 3;
            o[0] = c[0]; o[1] = c[1]; o[2] = c[2];
        }
    }
}

extern "C" void kernel_launch(void* const* d_in, const int* in_sizes, int n_in,
                              void* d_out, int out_size, void* d_ws, size_t ws_size,
                              hipStream_t stream) {
    const float* x  = (const float*)d_in[0];   // (B, N, 6) f32
    const float* cb = (const float*)d_in[1];   // (K, 6)   f32
    float* out      = (float*)d_out;           // (B, N, 6) f32

    const int P = in_sizes[0] / 6;
    const int K = in_sizes[1] / 6;

    const int tiles  = (P + TILE_P - 1) / TILE_P;
    int blocks = (tiles + WAVES_PER_BLOCK - 1) / WAVES_PER_BLOCK;
    if (blocks > MAX_BLOCKS) blocks = MAX_BLOCKS;

    bent_quant_wmma_kernel<<<blocks, BLOCK_THREADS, 0, stream>>>(x, cb, out, P, K);
}